// LocationRelativeAttention_22402549416652
// MI455X (gfx1250) — compile-verified
//
#include <hip/hip_runtime.h>
#include <math.h>

#define Bsz   256
#define Tlen  1024
#define Hdim  512
#define QHdim 1024
#define Kk    31
#define Wlen  256
#define PADc  15
#define CAW   (Tlen + 2 * PADc)   // 1054

typedef __attribute__((ext_vector_type(2))) float v2f;
typedef __attribute__((ext_vector_type(8))) float v8f;

// ---------------------------------------------------------------------------
// Kernel 1: q[b,h] = query[0,b,:] . Wq[h,:] + bq[h]
// One wave32 per 16(B) x 16(H) output tile, V_WMMA_F32_16X16X4_F32 over QH.
// A (16x4, M=B rows) per-lane: lanes 0-15 hold K={k0,k0+1}, lanes 16-31 K={k0+2,k0+3}.
// B (4x16, N=H cols) = Wq^T, same K interleave, N striped over lanes.
// C/D: VGPR r -> M = r (lanes 0-15) / r+8 (lanes 16-31), N = lane&15.
// ---------------------------------------------------------------------------
__global__ __launch_bounds__(32) void qproj_wmma_kernel(
    const float* __restrict__ Q,    // [B, QH]  (query[0])
    const float* __restrict__ Wq,   // [H, QH]
    const float* __restrict__ bq,   // [H]
    float* __restrict__ qbuf)       // [B, H]
{
    const int lane = threadIdx.x & 31;
    const int l16  = lane & 15;
    const int half = lane >> 4;
    const int m0   = blockIdx.x << 4;   // tile over B
    const int n0   = blockIdx.y << 4;   // tile over H

    const float* qrow = Q  + (size_t)(m0 + l16) * QHdim + 2 * half;
    const float* wrow = Wq + (size_t)(n0 + l16) * QHdim + 2 * half;

    v8f acc = {};
    #pragma unroll 8
    for (int k0 = 0; k0 < QHdim; k0 += 4) {
        v2f a  = *(const v2f*)(qrow + k0);
        v2f bm = *(const v2f*)(wrow + k0);
        acc = __builtin_amdgcn_wmma_f32_16x16x4_f32(
            /*neg_a=*/false, a, /*neg_b=*/false, bm,
            /*c_mod=*/(short)0, acc, /*reuse_a=*/false, /*reuse_b=*/false);
    }

    const float bqv = bq[n0 + l16];
    #pragma unroll
    for (int r = 0; r < 8; ++r) {
        const int m = m0 + r + 8 * half;
        qbuf[(size_t)m * Hdim + n0 + l16] = acc[r] + bqv;
    }
}

// ---------------------------------------------------------------------------
// Kernel 2: one workgroup (256 threads) per batch b.
//   conv_w (63.5 KB) is DMA'd global->LDS with GLOBAL_LOAD_ASYNC_TO_LDS_B128
//   (ASYNCcnt path, no VGPR round-trip), overlapped with the other staging.
//   thread w: score[w] = sum_h v[h]*tanh(conv(lf)[h,w] + q[b,h] + conv_b[h])
//   block softmax + first-index argmax, scatter outputs, weighted tokens gather.
// ---------------------------------------------------------------------------
__global__ __launch_bounds__(256) void attn_fused_kernel(
    const float* __restrict__ tokens,      // [T, B, H]
    const unsigned char* __restrict__ mask,// [B, T] (bool)
    const int* __restrict__ num_tokens,    // [B]
    const float* __restrict__ ca,          // [B, CAW]
    const int* __restrict__ wstart,        // [B]
    const float* __restrict__ conv_w,      // [H, 1, K] = [H*K]
    const float* __restrict__ conv_b,      // [H]
    const float* __restrict__ v,           // [H]
    const float* __restrict__ qbuf,        // [B, H]
    float* __restrict__ out)
{
    __shared__ float s_cw[Hdim * Kk];        // 63488 B
    __shared__ float s_lf[Wlen + 2 * PADc];  // 1144 B
    __shared__ float s_qc[Hdim];             // q + conv_b
    __shared__ float s_v[Hdim];
    __shared__ float s_red[256];
    __shared__ int   s_argm[256];
    __shared__ float s_align[Wlen];

    const int b   = blockIdx.x;
    const int tid = threadIdx.x;
    const int ws  = wstart[b];

    // --- async DMA: conv_w -> LDS, 16B per lane per issue (ASYNCcnt path) ---
    {
        const unsigned lds_base = (unsigned)(uintptr_t)(void*)s_cw; // LDS logical addr = flat[31:0]
        const unsigned long long gbase = (unsigned long long)(uintptr_t)conv_w;
        const int nchunk = (Hdim * Kk * 4) / 16;   // 3968 x 16B
        for (int c = tid; c < nchunk; c += 256) {
            unsigned            ldst = lds_base + (unsigned)(c * 16);
            unsigned long long  gsrc = gbase + (unsigned long long)(c * 16);
            asm volatile("global_load_async_to_lds_b128 %0, %1, off"
                         :: "v"(ldst), "v"(gsrc) : "memory");
        }
    }

    // --- overlap: stage the small arrays through VGPRs while DMA runs ---
    for (int j = tid; j < Wlen + 2 * PADc; j += 256) {
        const int g = ws + j;
        float val = ca[(size_t)b * CAW + g];
        if (g >= PADc && g < PADc + Tlen && !mask[(size_t)b * Tlen + (g - PADc)])
            val = 0.f;
        s_lf[j] = val;
    }
    for (int h = tid; h < Hdim; h += 256) {
        s_qc[h] = qbuf[(size_t)b * Hdim + h] + conv_b[h];
        s_v[h]  = v[h];
    }

    // drain async DMA, then make all LDS visible block-wide
    asm volatile("s_wait_asynccnt 0x0" ::: "memory");
    __syncthreads();

    // --- per-thread window position: conv + tanh + dot-with-v ---
    const int w = tid;
    float lfr[Kk];
    #pragma unroll
    for (int k = 0; k < Kk; ++k) lfr[k] = s_lf[w + k];

    float score = 0.f;
    for (int h = 0; h < Hdim; ++h) {
        const float* cwh = &s_cw[h * Kk];   // uniform addr -> LDS broadcast
        float c = 0.f;
        #pragma unroll
        for (int k = 0; k < Kk; ++k) c = fmaf(cwh[k], lfr[k], c);
        score += s_v[h] * tanhf(c + s_qc[h]);
    }
    if (!mask[(size_t)b * Tlen + ws + w]) score = -__builtin_inff();

    // --- softmax over the 256 window positions ---
    s_red[tid] = score; __syncthreads();
    for (int st = 128; st > 0; st >>= 1) {
        if (tid < st) s_red[tid] = fmaxf(s_red[tid], s_red[tid + st]);
        __syncthreads();
    }
    const float mx = s_red[0]; __syncthreads();
    const float e = __expf(score - mx);
    s_red[tid] = e; __syncthreads();
    for (int st = 128; st > 0; st >>= 1) {
        if (tid < st) s_red[tid] += s_red[tid + st];
        __syncthreads();
    }
    const float al = e * (1.f / s_red[0]);
    s_align[w] = al;
    // first-index argmax: score == mx is exact (mx came from these values)
    s_argm[tid] = (score == mx) ? tid : 0x7FFFFFFF;
    __syncthreads();
    for (int st = 128; st > 0; st >>= 1) {
        if (tid < st) s_argm[tid] = min(s_argm[tid], s_argm[tid + st]);
        __syncthreads();
    }

    // --- output pointers (flat concatenation, reference return order) ---
    float* ctx_out   = out;                                  // [B, H]
    float* align_out = out + (size_t)Bsz * Hdim;             // [B, T]
    float* cap_out   = align_out + (size_t)Bsz * Tlen;       // [B, CAW]
    int*   ws_out    = (int*)(cap_out + (size_t)Bsz * CAW);  // [B] int32 bits

    // align_full[:, pad:T+pad] -> window positions hold align, rest 0
    for (int j = tid; j < Tlen; j += 256) {
        const int d = j - ws;
        align_out[(size_t)b * Tlen + j] = (d >= 0 && d < Wlen) ? s_align[d] : 0.f;
    }
    // ca + align_full over the padded axis
    for (int g = tid; g < CAW; g += 256) {
        float val = ca[(size_t)b * CAW + g];
        if (g >= PADc && g < PADc + Tlen && !mask[(size_t)b * Tlen + (g - PADc)])
            val = 0.f;
        const int d = g - PADc - ws;
        if (d >= 0 && d < Wlen) val += s_align[d];
        cap_out[(size_t)b * CAW + g] = val;
    }
    // ws_new = clip(min(ws + argmax - W/2, num_tokens - W), 0)
    if (tid == 0) {
        const int wi = s_argm[0];
        int wn;
        if (wi == 0x7FFFFFFF) {
            wn = 0;   // degenerate all -inf -> argmax 0 -> clipped to 0
        } else {
            wn = ws + wi - (Wlen / 2);
            wn = min(wn, num_tokens[b] - Wlen);
            wn = max(wn, 0);
        }
        ws_out[b] = wn;
    }

    // --- context = sum_w align[w] * tokens[ws+w, b, :]  (coalesced over H) ---
    float c0 = 0.f, c1 = 0.f;
    for (int wi = 0; wi < Wlen; ++wi) {
        if (wi + 8 < Wlen)
            __builtin_prefetch(
                tokens + ((size_t)(ws + wi + 8) * Bsz + b) * Hdim + tid, 0, 1);
        const float a = s_align[wi];
        const float* tok = tokens + ((size_t)(ws + wi) * Bsz + b) * Hdim;
        c0 = fmaf(a, tok[tid], c0);
        c1 = fmaf(a, tok[tid + 256], c1);
    }
    ctx_out[(size_t)b * Hdim + tid]       = c0;
    ctx_out[(size_t)b * Hdim + tid + 256] = c1;
}

// ---------------------------------------------------------------------------
extern "C" void kernel_launch(void* const* d_in, const int* in_sizes, int n_in,
                              void* d_out, int out_size, void* d_ws, size_t ws_size,
                              hipStream_t stream) {
    const float*         tokens  = (const float*)d_in[0];
    const unsigned char* mask    = (const unsigned char*)d_in[1];
    const int*           ntok    = (const int*)d_in[2];
    const float*         query   = (const float*)d_in[3];   // [1,B,QH]
    const float*         ca      = (const float*)d_in[4];
    const int*           wstart  = (const int*)d_in[5];
    const float*         conv_w  = (const float*)d_in[6];
    const float*         conv_b  = (const float*)d_in[7];
    const float*         Wq      = (const float*)d_in[8];
    const float*         bq      = (const float*)d_in[9];
    const float*         v       = (const float*)d_in[10];
    float* out  = (float*)d_out;
    float* qbuf = (float*)d_ws;   // [B, H] = 512 KB scratch

    dim3 g1(Bsz / 16, Hdim / 16);   // 16 x 32 tiles, 1 wave each
    qproj_wmma_kernel<<<g1, 32, 0, stream>>>(query, Wq, bq, qbuf);

    attn_fused_kernel<<<Bsz, 256, 0, stream>>>(
        tokens, mask, ntok, ca, wstart, conv_w, conv_b, v, qbuf, out);
}